// TTFeedForward_81217831567480
// MI455X (gfx1250) — compile-verified
//
#include <hip/hip_runtime.h>

// ---------------------------------------------------------------------------
// TT feed-forward (fc1 + exact GELU + fc2) for MI455X / gfx1250.
// All heavy contractions run on V_WMMA_F32_16X16X4_F32 out of LDS; the
// 256x256 middle TT-cores are staged into LDS by the Tensor Data Mover
// (tensor_load_to_lds with LDS padding), waited via s_wait_tensorcnt.
// ---------------------------------------------------------------------------

typedef float v2f __attribute__((ext_vector_type(2)));
typedef float v8f __attribute__((ext_vector_type(8)));
typedef unsigned int u32x4 __attribute__((ext_vector_type(4)));
typedef int i32x4 __attribute__((ext_vector_type(4)));
typedef int i32x8 __attribute__((ext_vector_type(8)));

#if __has_builtin(__builtin_amdgcn_tensor_load_to_lds) && __has_builtin(__builtin_amdgcn_s_wait_tensorcnt)
#define TT_HAS_TDM 1
#else
#define TT_HAS_TDM 0
#endif

__device__ __forceinline__ v8f wmma_f32(v2f a, v2f b, v8f c) {
  // D = A(16x4,f32) * B(4x16,f32) + C(16x16,f32)
  return __builtin_amdgcn_wmma_f32_16x16x4_f32(false, a, false, b, (short)0, c,
                                               false, false);
}

__device__ __forceinline__ float gelu_exact(float x) {
  return 0.5f * x * (1.0f + erff(x * 0.70710678118654752440f));
}

// Stage a 256x256 f32 matrix from global into LDS with row stride 257 floats
// (bank-conflict pad). TDM path: pad_enable adds 1 dword after every 256
// dwords stored. Fallback: cooperative copy.
__device__ __forceinline__ void stage_w_256x256(float* lds_dst, const float* gsrc,
                                                unsigned lds_byte_off, int tid) {
#if TT_HAS_TDM
  if (tid < 32) {  // one wave issues the DMA (EXEC is ignored by TDM)
    unsigned long long ga = (unsigned long long)gsrc;
    u32x4 g0;
    g0.x = 1u;                                   // count=1, user descriptor
    g0.y = lds_byte_off;                         // lds_addr (bytes)
    g0.z = (unsigned)ga;                         // global_addr[31:0]
    g0.w = (unsigned)((ga >> 32) & 0x01FFFFFFu)  // global_addr[56:32]
         | (2u << 30);                           // type = 2 ("image")
    i32x8 g1;
    g1[0] = (2 << 16)        // data_size = 4 bytes
          | (1 << 20)        // pad_enable
          | (7 << 22);       // pad_interval = 256 dwords (pad_amount=0 -> 1 dw)
    g1[1] = (int)(256u << 16);  // tensor_dim0 = 256
    g1[2] = (int)(256u << 16);  // tensor_dim1 = 256
    g1[3] = (int)(256u << 16);  // tile_dim0   = 256
    g1[4] = 256;                // tile_dim1   = 256
    g1[5] = 256;                // tensor_dim0_stride = 256
    g1[6] = 0;
    g1[7] = 0;
    i32x4 gz; gz[0] = 0; gz[1] = 0; gz[2] = 0; gz[3] = 0;
    i32x8 gz8; gz8[0] = 0; gz8[1] = 0; gz8[2] = 0; gz8[3] = 0;
    gz8[4] = 0; gz8[5] = 0; gz8[6] = 0; gz8[7] = 0;
    // 6-arg form (this toolchain): (g0, g1, g2, g3, g4, cpol)
    __builtin_amdgcn_tensor_load_to_lds(g0, g1, gz, gz, gz8, 0);
  }
  (void)lds_dst;
#else
  for (int i = tid; i < 256 * 256; i += 256)
    lds_dst[(i >> 8) * 257 + (i & 255)] = gsrc[i];
  (void)lds_byte_off;
#endif
}

__device__ __forceinline__ void stage_wait(int tid) {
#if TT_HAS_TDM
  if (tid < 32) __builtin_amdgcn_s_wait_tensorcnt(0);
#endif
  __syncthreads();
}

// ---------------------------------------------------------------------------
// Kernel A: fc1 (1024 -> 4096) + exact GELU.  2 samples per workgroup.
// LDS layout (float offsets):
#define A_W1 0        // 256x257  c1 as W1[(q1,n2)][(m2,q2)], padded
#define A_W0 65792    // 8x256    c0 as W0[n1][(m1,q1)]
#define A_W2 67840    // 128x16   c2 as W2[(q2,n3)][m3]
#define A_X  69888    // 2x1024   input rows
#define A_Y1 71936    // 2x128x17 Y1[s][(n2,n3)][q1] (current m1)
#define A_Y2 76288    // 16x257   Y2[(s*8+n3)][(m2,q2)]
#define A_ZA 80400    // 2x256    z-slice accumulator
#define A_TOT 80912   // 323,648 bytes < 320 KB
// ---------------------------------------------------------------------------
__global__ __launch_bounds__(256) void tt_fc1_gelu_kernel(
    const float* __restrict__ x, const float* __restrict__ c0,
    const float* __restrict__ c1, const float* __restrict__ c2,
    const float* __restrict__ bias, float* __restrict__ h) {
  __shared__ float sm[A_TOT];
  const int tid = threadIdx.x;
  const int lane = tid & 31;
  const int wv = tid >> 5;
  const int s0 = blockIdx.x * 2;

  stage_w_256x256(&sm[A_W1], c1, (unsigned)(A_W1 * 4), tid);
  for (int i = tid; i < 2048; i += 256) sm[A_W0 + i] = c0[i];
  for (int i = tid; i < 2048; i += 256) sm[A_W2 + i] = c2[i];
  for (int i = tid; i < 2048; i += 256) sm[A_X + i] = x[(size_t)s0 * 1024 + i];
  stage_wait(tid);

  for (int m1 = 0; m1 < 16; ++m1) {
    // -- zero z-accumulator; step 1 (VALU): Y1[s][(n2,n3)][q1] over n1 --
    sm[A_ZA + tid] = 0.0f;
    sm[A_ZA + 256 + tid] = 0.0f;
    for (int t = tid; t < 4096; t += 256) {
      const int s = t >> 11;
      const int r = (t >> 4) & 127;  // n2*8 + n3
      const int q1 = t & 15;
      float acc = 0.0f;
#pragma unroll
      for (int n1 = 0; n1 < 8; ++n1)
        acc += sm[A_X + s * 1024 + n1 * 128 + r] *
               sm[A_W0 + n1 * 256 + m1 * 16 + q1];
      sm[A_Y1 + (s * 128 + r) * 17 + q1] = acc;
    }
    __syncthreads();

    // -- step 2 (WMMA): M=(s,n3)=16, K=(q1,n2)=256, N=(m2,q2)=256 --
    {
      const int arow = lane & 15;        // M row
      const int s = arow >> 3;
      const int n3 = arow & 7;
      const int khalf = (lane >> 4) << 1;
      const int j0 = wv * 2;             // this wave's two N-tiles
      const int ncol = lane & 15;
      const int y1b = A_Y1 + s * (128 * 17) + n3 * 17;
      v8f acc0 = {}; v8f acc1 = {};
      for (int kk = 0; kk < 64; ++kk) {
        const int ka = kk * 4 + khalf;
        const int kb = ka + 1;
        v2f a;
        a.x = sm[y1b + (ka & 15) * (8 * 17) + (ka >> 4)];
        a.y = sm[y1b + (kb & 15) * (8 * 17) + (kb >> 4)];
        v2f b0, b1;
        b0.x = sm[A_W1 + ka * 257 + j0 * 16 + ncol];
        b0.y = sm[A_W1 + kb * 257 + j0 * 16 + ncol];
        b1.x = sm[A_W1 + ka * 257 + j0 * 16 + 16 + ncol];
        b1.y = sm[A_W1 + kb * 257 + j0 * 16 + 16 + ncol];
        acc0 = wmma_f32(a, b0, acc0);
        acc1 = wmma_f32(a, b1, acc1);
      }
#pragma unroll
      for (int r = 0; r < 8; ++r) {
        const int m = r + ((lane >> 4) << 3);
        sm[A_Y2 + m * 257 + j0 * 16 + ncol] = acc0[r];
        sm[A_Y2 + m * 257 + j0 * 16 + 16 + ncol] = acc1[r];
      }
    }
    __syncthreads();

    // -- step 3 (WMMA): per sample M=m2=16, K=(q2,n3)=128, N=m3=16;
    //    K split over 4 waves per sample, LDS float-atomic reduce --
    {
      const int s = wv >> 2;
      const int kofs = (wv & 3) * 32;
      const int khalf = (lane >> 4) << 1;
      const int arow = lane & 15;  // m2
      const int ncol = lane & 15;  // m3
      v8f acc = {};
#pragma unroll
      for (int kk = 0; kk < 8; ++kk) {
        const int ka = kofs + kk * 4 + khalf;
        const int kb = ka + 1;
        v2f a;
        a.x = sm[A_Y2 + (s * 8 + (ka & 7)) * 257 + arow * 16 + (ka >> 3)];
        a.y = sm[A_Y2 + (s * 8 + (kb & 7)) * 257 + arow * 16 + (kb >> 3)];
        v2f b;
        b.x = sm[A_W2 + ka * 16 + ncol];
        b.y = sm[A_W2 + kb * 16 + ncol];
        acc = wmma_f32(a, b, acc);
      }
#pragma unroll
      for (int r = 0; r < 8; ++r) {
        const int m2 = r + ((lane >> 4) << 3);
        atomicAdd(&sm[A_ZA + s * 256 + m2 * 16 + ncol], acc[r]);
      }
    }
    __syncthreads();

    // -- bias + exact GELU + store h slice --
    for (int t = tid; t < 512; t += 256) {
      const int s = t >> 8;
      const int idx = t & 255;
      const float z = sm[A_ZA + t] + bias[m1 * 256 + idx];
      h[(size_t)(s0 + s) * 4096 + m1 * 256 + idx] = gelu_exact(z);
    }
    __syncthreads();
  }
}

// ---------------------------------------------------------------------------
// Kernel B: fc2 (4096 -> 1024).  1 sample per workgroup.
#define B_V1 0        // 256x257  fc2_c1 as V1[(q1,m2)][(u2,q2)], padded
#define B_V0 65792    // 16x128   fc2_c0 as V0[m1][(u1,q1)]
#define B_V2 67840    // 256x16   fc2_c2 as V2[(q2,m3)][u3], cols 8..15 zero
#define B_Y1 71936    // 256x17   Y1[(m2*16+m3)][q1] (current u1)
#define B_Y2 76288    // 16x257   Y2[m3][(u2,q2)]
#define B_ZB 80400    // 256
#define B_TOT 80656   // 322,624 bytes < 320 KB
// ---------------------------------------------------------------------------
__global__ __launch_bounds__(256) void tt_fc2_kernel(
    const float* __restrict__ h, const float* __restrict__ c0,
    const float* __restrict__ c1, const float* __restrict__ c2,
    const float* __restrict__ bias, float* __restrict__ out) {
  __shared__ float sm[B_TOT];
  const int tid = threadIdx.x;
  const int lane = tid & 31;
  const int wv = tid >> 5;
  const int s = blockIdx.x;

  stage_w_256x256(&sm[B_V1], c1, (unsigned)(B_V1 * 4), tid);
  for (int i = tid; i < 2048; i += 256) sm[B_V0 + i] = c0[i];
  for (int i = tid; i < 4096; i += 256) {  // zero-pad u3 to 16 columns
    const int rr = i >> 4, cc = i & 15;
    sm[B_V2 + i] = (cc < 8) ? c2[rr * 8 + cc] : 0.0f;
  }
  stage_wait(tid);

  // thread tid owns h row (m2,m3)=(tid>>4, tid&15); hoist h loads out of u1 loop
  float hv[16];
#pragma unroll
  for (int m1 = 0; m1 < 16; ++m1)
    hv[m1] = h[(size_t)s * 4096 + m1 * 256 + tid];

  for (int u1 = 0; u1 < 8; ++u1) {
    sm[B_ZB + tid] = 0.0f;  // zero accumulator
    // -- step 1 (VALU): Y1[(m2,m3)][q1] over m1 --
#pragma unroll
    for (int q1 = 0; q1 < 16; ++q1) {
      float acc = 0.0f;
#pragma unroll
      for (int m1 = 0; m1 < 16; ++m1)
        acc += hv[m1] * sm[B_V0 + m1 * 128 + u1 * 16 + q1];
      sm[B_Y1 + tid * 17 + q1] = acc;
    }
    __syncthreads();

    // -- step 2 (WMMA): M=m3=16, K=(q1,m2)=256, N=(u2,q2)=256 --
    {
      const int arow = lane & 15;  // m3
      const int khalf = (lane >> 4) << 1;
      const int j0 = wv * 2;
      const int ncol = lane & 15;
      v8f acc0 = {}; v8f acc1 = {};
      for (int kk = 0; kk < 64; ++kk) {
        const int ka = kk * 4 + khalf;
        const int kb = ka + 1;
        v2f a;
        a.x = sm[B_Y1 + ((ka & 15) * 16 + arow) * 17 + (ka >> 4)];
        a.y = sm[B_Y1 + ((kb & 15) * 16 + arow) * 17 + (kb >> 4)];
        v2f b0, b1;
        b0.x = sm[B_V1 + ka * 257 + j0 * 16 + ncol];
        b0.y = sm[B_V1 + kb * 257 + j0 * 16 + ncol];
        b1.x = sm[B_V1 + ka * 257 + j0 * 16 + 16 + ncol];
        b1.y = sm[B_V1 + kb * 257 + j0 * 16 + 16 + ncol];
        acc0 = wmma_f32(a, b0, acc0);
        acc1 = wmma_f32(a, b1, acc1);
      }
#pragma unroll
      for (int r = 0; r < 8; ++r) {
        const int m = r + ((lane >> 4) << 3);  // m3
        sm[B_Y2 + m * 257 + j0 * 16 + ncol] = acc0[r];
        sm[B_Y2 + m * 257 + j0 * 16 + 16 + ncol] = acc1[r];
      }
    }
    __syncthreads();

    // -- step 3 (WMMA): M=u2=16, K=(q2,m3)=256 split over 8 waves, N=u3(pad 16)
    {
      const int kofs = wv * 32;
      const int khalf = (lane >> 4) << 1;
      const int arow = lane & 15;  // u2
      const int ncol = lane & 15;  // u3 (padded)
      v8f acc = {};
#pragma unroll
      for (int kk = 0; kk < 8; ++kk) {
        const int ka = kofs + kk * 4 + khalf;
        const int kb = ka + 1;
        v2f a;
        a.x = sm[B_Y2 + (ka & 15) * 257 + arow * 16 + (ka >> 4)];
        a.y = sm[B_Y2 + (kb & 15) * 257 + arow * 16 + (kb >> 4)];
        v2f b;
        b.x = sm[B_V2 + ka * 16 + ncol];
        b.y = sm[B_V2 + kb * 16 + ncol];
        acc = wmma_f32(a, b, acc);
      }
#pragma unroll
      for (int r = 0; r < 8; ++r) {
        const int u2 = r + ((lane >> 4) << 3);
        atomicAdd(&sm[B_ZB + u2 * 16 + ncol], acc[r]);
      }
    }
    __syncthreads();

    if (tid < 128) {  // t = u2*8+u3
      const int u2 = tid >> 3, u3 = tid & 7;
      out[(size_t)s * 1024 + u1 * 128 + tid] =
          sm[B_ZB + u2 * 16 + u3] + bias[u1 * 128 + tid];
    }
    __syncthreads();
  }
}

// ---------------------------------------------------------------------------
extern "C" void kernel_launch(void* const* d_in, const int* in_sizes, int n_in,
                              void* d_out, int out_size, void* d_ws, size_t ws_size,
                              hipStream_t stream) {
  (void)in_sizes; (void)n_in; (void)out_size; (void)ws_size;
  const float* x   = (const float*)d_in[0];
  const float* f10 = (const float*)d_in[1];
  const float* f11 = (const float*)d_in[2];
  const float* f12 = (const float*)d_in[3];
  const float* f1b = (const float*)d_in[4];
  const float* f20 = (const float*)d_in[5];
  const float* f21 = (const float*)d_in[6];
  const float* f22 = (const float*)d_in[7];
  const float* f2b = (const float*)d_in[8];
  float* outp = (float*)d_out;
  float* hbuf = (float*)d_ws;  // 4096 x 4096 f32 = 64 MB scratch

  tt_fc1_gelu_kernel<<<2048, 256, 0, stream>>>(x, f10, f11, f12, f1b, hbuf);
  tt_fc2_kernel<<<4096, 256, 0, stream>>>(hbuf, f20, f21, f22, f2b, outp);
}